// MultiBoxLoss_81707457839596
// MI455X (gfx1250) — compile-verified
//
#include <hip/hip_runtime.h>
#include <hip/hip_bf16.h>
#include <math.h>

#define PRIORS 24564
#define BATCH  128
#define NPROP  100
#define CLS    21
#define BT     1024   // threads per block, kernel A (32 waves)
#define NTILE  ((PRIORS + BT - 1) / BT)   // 24 tiles, last one partial (1012)

typedef float v2f __attribute__((ext_vector_type(2)));
typedef float v4f __attribute__((ext_vector_type(4)));
typedef float v8f __attribute__((ext_vector_type(8)));

// ---------------------------------------------------------------------------
// CDNA5 async global->LDS copies (ASYNCcnt-tracked, CDNA5 ISA).
// GVS addressing: mem = SADDR(64b sgpr) + VADDR(32b vgpr byte offset).
// VDST vgpr holds the LDS byte offset (low 32 bits of the flat shared address).
// ---------------------------------------------------------------------------
__device__ __forceinline__ unsigned lds_off_of(const void* p) {
  return (unsigned)(uintptr_t)p;   // flat->LDS aperture: addr[31:0] is LDS offset
}
__device__ __forceinline__ void async_ld_b32(unsigned lds, unsigned voff,
                                             unsigned long long base) {
  asm volatile("global_load_async_to_lds_b32 %0, %1, %2"
               :: "v"(lds), "v"(voff), "s"(base) : "memory");
}
__device__ __forceinline__ void async_ld_b64(unsigned lds, unsigned voff,
                                             unsigned long long base) {
  asm volatile("global_load_async_to_lds_b64 %0, %1, %2"
               :: "v"(lds), "v"(voff), "s"(base) : "memory");
}

// Wave32 sum reduction on the matrix pipe (V_WMMA_F32_16X16X4_F32 with B=ones):
// D[m][n] = x[m] + x[m+16]; 8 local adds cover m=0..7 / 8..15, one xor-16
// shuffle completes the 32-lane sum.
__device__ __forceinline__ float wave_sum_wmma(float x) {
  v2f a; a.x = x;    a.y = 0.0f;
  v2f b; b.x = 1.0f; b.y = 1.0f;
  v8f c = {};
  c = __builtin_amdgcn_wmma_f32_16x16x4_f32(false, a, false, b, (short)0, c,
                                            false, false);
  float t = ((c.s0 + c.s1) + (c.s2 + c.s3)) + ((c.s4 + c.s5) + (c.s6 + c.s7));
  t += __shfl_xor(t, 16, 32);
  return t;
}

// ---------------------------------------------------------------------------
// Kernel A: conf branch + smooth-L1. One workgroup per batch row; the whole
// row of mined CE values stays resident in the 320 KB WGP LDS, and the
// streaming inputs are staged through double-buffered async-to-LDS tiles.
// ---------------------------------------------------------------------------
__global__ void __launch_bounds__(BT) conf_branch_kernel(
    const float* __restrict__ loc_data, const float* __restrict__ conf_data,
    const float* __restrict__ loc_t, const int* __restrict__ conf_t,
    float* __restrict__ gacc) {
  extern __shared__ char smem_raw[];
  float*    mined   = (float*)smem_raw;                // PRIORS floats (98256 B)
  unsigned* hist    = (unsigned*)(mined + PRIORS);     // 256 bins
  float*    facc    = (float*)(hist + 256);            // [0]=loss_l [1]=sum_pos_ce [2]=sum_gt
  int*      iacc    = (int*)(facc + 8);                // [0]=npos [1]=sel [2]=kp
  int*      ct_tile = (int*)(iacc + 8);                // 2 * BT ints   (8 KB)
  v2f*      cf_tile = (v2f*)(ct_tile + 2 * BT);        // 2 * BT v2f   (16 KB)

  const int b = blockIdx.x;
  const int tid = threadIdx.x;
  if (tid < 8) { facc[tid] = 0.0f; iacc[tid] = 0; }
  __syncthreads();

  const v4f* loc4  = (const v4f*)loc_data + (size_t)b * PRIORS;
  const v4f* loct4 = (const v4f*)loc_t    + (size_t)b * PRIORS;
  const unsigned long long ct_base =
      (unsigned long long)(uintptr_t)(conf_t + (size_t)b * PRIORS);
  const unsigned long long cf_base =
      (unsigned long long)(uintptr_t)(conf_data + (size_t)b * PRIORS * 2);

  const unsigned ct_lds = lds_off_of(ct_tile) + (unsigned)tid * 4u;
  const unsigned cf_lds = lds_off_of(cf_tile) + (unsigned)tid * 8u;

  // ---- Phase 1: async double-buffered stream over conf_t / conf_data ----
  // Prologue: tile 0 in flight (every wave issues exactly 2 async ops/tile).
  async_ld_b32(ct_lds, (unsigned)tid * 4u, ct_base);
  async_ld_b64(cf_lds, (unsigned)tid * 8u, cf_base);

  float loss_l_p = 0.0f, sum_pos_p = 0.0f;
  int   npos_p = 0;
  for (int t = 0; t < NTILE; ++t) {
    const int buf = t & 1;
    if (t + 1 < NTILE) {
      const int nb = buf ^ 1;
      const int inext = (t + 1) * BT + tid;
      if (inext < PRIORS) {   // partial last tile: every wave still has lanes
        async_ld_b32(ct_lds + (unsigned)(nb * BT * 4), (unsigned)inext * 4u, ct_base);
        async_ld_b64(cf_lds + (unsigned)(nb * BT * 8), (unsigned)inext * 8u, cf_base);
      }
      // 2 newer ops may stay outstanding; the 2 oldest (current tile) landed.
      asm volatile("s_wait_asynccnt 0x2" ::: "memory");
    } else {
      asm volatile("s_wait_asynccnt 0x0" ::: "memory");
    }

    const int i = t * BT + tid;
    if (i < PRIORS) {
      int  tv  = ct_tile[buf * BT + tid];
      v2f  cv  = cf_tile[buf * BT + tid];
      bool pos = tv > 0;
      float m   = fmaxf(cv.x, cv.y);
      float lse = m + __logf(__expf(cv.x - m) + __expf(cv.y - m));
      float per = fmaxf(lse - (pos ? cv.y : cv.x), 0.0f);   // CE >= 0
      mined[i] = pos ? 0.0f : per;                          // loss_c[pos]=0 pre-rank
      if (pos) {
        npos_p++; sum_pos_p += per;
        v4f ld = __builtin_nontemporal_load(loc4 + i);
        v4f lt = __builtin_nontemporal_load(loct4 + i);
        #pragma unroll
        for (int c2 = 0; c2 < 4; ++c2) {
          float d  = ld[c2] - lt[c2];
          float ad = fabsf(d);
          loss_l_p += (ad < 1.0f) ? (0.5f * d * d) : (ad - 0.5f);
        }
      }
    }
  }
  // cross-lane reduction on the matrix pipe, then LDS atomics across waves
  float wl = wave_sum_wmma(loss_l_p);
  float wp = wave_sum_wmma(sum_pos_p);
  if ((tid & 31) == 0) { atomicAdd(&facc[0], wl); atomicAdd(&facc[1], wp); }
  if (npos_p) atomicAdd(&iacc[0], npos_p);
  __syncthreads();

  const int npos = iacc[0];
  int k = 3 * npos; if (k > PRIORS - 1) k = PRIORS - 1;

  float row_loss_c = 0.0f;
  if (k > 0) {
    // ---- Phase 2: 4-pass radix select for the k-th largest mined value ----
    // mined >= 0, so raw float bits are order-preserving.
    unsigned prefix = 0; int kp = k;
    for (int pass = 0; pass < 4; ++pass) {
      const int shift = 24 - pass * 8;
      if (tid < 256) hist[tid] = 0u;
      __syncthreads();
      const unsigned pmask = (pass == 0) ? 0u : (0xFFFFFFFFu << (shift + 8));
      for (int i = tid; i < PRIORS; i += BT) {
        unsigned key = __float_as_uint(mined[i]);
        if ((key & pmask) == prefix)
          atomicAdd(&hist[(key >> shift) & 255u], 1u);
      }
      __syncthreads();
      if (tid == 0) {
        int cum = 0, sel = 0;
        for (int d = 255; d >= 0; --d) {
          cum += (int)hist[d];
          if (cum >= kp) { sel = d; break; }
        }
        iacc[1] = sel;
        iacc[2] = kp - (cum - (int)hist[sel]);  // remaining within equal-digit group
      }
      __syncthreads();
      prefix |= ((unsigned)iacc[1]) << shift;
      kp = iacc[2];
    }
    const float T = __uint_as_float(prefix);   // threshold; kp = k - count(v > T)

    // ---- Phase 3: sum selected negatives (ties collapse to T * kp) ----
    float sgt = 0.0f;
    for (int i = tid; i < PRIORS; i += BT) {
      float v = mined[i];
      sgt += (v > T) ? v : 0.0f;
    }
    float wsgt = wave_sum_wmma(sgt);
    if ((tid & 31) == 0) atomicAdd(&facc[2], wsgt);
    __syncthreads();
    row_loss_c = facc[1] + facc[2] + T * (float)kp;
  }

  if (tid == 0) {
    atomicAdd(&gacc[0], facc[0]);       // loss_l (sum)
    atomicAdd(&gacc[1], row_loss_c);    // loss_c (sum)
    atomicAdd(&gacc[2], (float)npos);   // N
  }
}

// ---------------------------------------------------------------------------
// Kernel B: 21-way classifier branch over 100 proposals per row (tiny).
// Exact stable-rank selection via O(NPROP^2) comparisons in LDS.
// ---------------------------------------------------------------------------
__global__ void __launch_bounds__(128) cls_branch_kernel(
    const float* __restrict__ cls_data, const int* __restrict__ cls_t,
    float* __restrict__ gacc) {
  __shared__ float mined[NPROP];
  __shared__ float pers[NPROP];
  __shared__ int   posf[NPROP];
  __shared__ int   np_s;
  __shared__ float fsum;
  const int b = blockIdx.x, tid = threadIdx.x;
  if (tid == 0) { np_s = 0; fsum = 0.0f; }
  __syncthreads();

  if (tid < NPROP) {
    const float* row = cls_data + ((size_t)b * NPROP + tid) * CLS;
    int tv = cls_t[(size_t)b * NPROP + tid];
    float m = row[0];
    #pragma unroll
    for (int c2 = 1; c2 < CLS; ++c2) m = fmaxf(m, row[c2]);
    float s = 0.0f;
    #pragma unroll
    for (int c2 = 0; c2 < CLS; ++c2) s += __expf(row[c2] - m);
    float lse = m + __logf(s);
    float per = fmaxf(lse - row[tv], 0.0f);
    int pos = tv > 0;
    pers[tid]  = per;
    mined[tid] = pos ? 0.0f : per;
    posf[tid]  = pos;
    if (pos) atomicAdd(&np_s, 1);
  }
  __syncthreads();

  const int np = np_s;
  int k = 3 * np; if (k > NPROP - 1) k = NPROP - 1;

  float contrib = 0.0f;
  if (tid < NPROP) {
    if (posf[tid]) {
      contrib = pers[tid];
    } else {
      float v = mined[tid];
      int c = 0;
      for (int j = 0; j < NPROP; ++j) {
        float u = mined[j];
        c += (u > v) || ((u == v) && (j < tid));   // stable descending rank
      }
      if (c < k) contrib = v;
    }
  }
  float w = wave_sum_wmma(contrib);
  if ((tid & 31) == 0) atomicAdd(&fsum, w);
  __syncthreads();
  if (tid == 0) { atomicAdd(&gacc[3], fsum); atomicAdd(&gacc[4], (float)np); }
}

// ---------------------------------------------------------------------------
__global__ void zero_ws_kernel(float* __restrict__ ws) {
  if (threadIdx.x < 8) ws[threadIdx.x] = 0.0f;
}

__global__ void finalize_kernel(const float* __restrict__ ws,
                                float* __restrict__ out) {
  if (threadIdx.x == 0) {
    float N  = ws[2];
    float Nc = ws[4];
    out[0] = ws[0] / N;    // loss_l / N
    out[1] = ws[1] / N;    // loss_c / N
    out[2] = ws[3] / Nc;   // loss_cls / N_cls
  }
}

extern "C" void kernel_launch(void* const* d_in, const int* in_sizes, int n_in,
                              void* d_out, int out_size, void* d_ws, size_t ws_size,
                              hipStream_t stream) {
  const float* loc_data  = (const float*)d_in[0];
  const float* conf_data = (const float*)d_in[1];
  const float* cls_data  = (const float*)d_in[2];
  const float* loc_t     = (const float*)d_in[3];
  const int*   conf_t    = (const int*)d_in[4];
  const int*   cls_t     = (const int*)d_in[5];
  float* out = (float*)d_out;
  float* ws  = (float*)d_ws;

  const size_t smem = (size_t)PRIORS * 4 + 256 * 4 + 8 * 4 + 8 * 4
                    + (size_t)2 * BT * 4      // ct_tile double buffer
                    + (size_t)2 * BT * 8;     // cf_tile double buffer  (~124 KB)
  (void)hipFuncSetAttribute(reinterpret_cast<const void*>(conf_branch_kernel),
                            hipFuncAttributeMaxDynamicSharedMemorySize,
                            (int)smem);

  zero_ws_kernel<<<1, 32, 0, stream>>>(ws);
  conf_branch_kernel<<<BATCH, BT, smem, stream>>>(loc_data, conf_data, loc_t,
                                                  conf_t, ws);
  cls_branch_kernel<<<BATCH, 128, 0, stream>>>(cls_data, cls_t, ws);
  finalize_kernel<<<1, 32, 0, stream>>>(ws, out);
}